// QiskitEmotionClassifier_70377334112281
// MI455X (gfx1250) — compile-verified
//
#include <hip/hip_runtime.h>
#include <math.h>

typedef float v2f __attribute__((ext_vector_type(2)));
typedef float v8f __attribute__((ext_vector_type(8)));

#define DIM 64

// D = A(16x4) * B(4x16) + C, all f32 (exact fp32 WMMA path on gfx1250)
__device__ __forceinline__ v8f wmma_f32_k4(v2f a, v2f b, v8f c) {
  return __builtin_amdgcn_wmma_f32_16x16x4_f32(false, a, false, b, (short)0, c, false, false);
}

struct c2 { float r, i; };
__device__ __forceinline__ c2 cmul(c2 a, c2 b){ return { a.r*b.r - a.i*b.i, a.r*b.i + a.i*b.r }; }
__device__ __forceinline__ c2 cadd(c2 a, c2 b){ return { a.r+b.r, a.i+b.i }; }

__device__ __forceinline__ int imin(int a, int b){ return a < b ? a : b; }

__global__ __launch_bounds__(256)
void qnn_fused(const float* __restrict__ x,
               const float* __restrict__ pre_w1, const float* __restrict__ pre_b1,
               const float* __restrict__ pre_w2, const float* __restrict__ pre_b2,
               const float* __restrict__ qw,
               const float* __restrict__ post_w1, const float* __restrict__ post_b1,
               const float* __restrict__ post_w2, const float* __restrict__ post_b2,
               float* __restrict__ out)
{
  __shared__ float ldsH[8][2][256];   // per-wave 16x16 activation tiles (2 tiles)
  __shared__ float ldsV[8][32][6];    // per-wave angles / expectations staging

  const int tid  = threadIdx.x;
  const int lane = tid & 31;
  const int wv   = tid >> 5;
  const int half = lane >> 4;   // lane half -> K/M split in WMMA layouts
  const int ln   = lane & 15;
  const int ln6  = imin(ln, 5);       // clamped (in-bounds) indices; select 0 after load
  const int ln4  = imin(ln, 3);
  const long base = (long)blockIdx.x * 256 + (long)wv * 32;

  // ---------------- P1: h = relu(x @ pre_w1 + pre_b1) via WMMA ----------------
  {
    v2f b1;
    b1.x = pre_w1[(half*2 + 0)*16 + ln];
    b1.y = pre_w1[(half*2 + 1)*16 + ln];
    float bias = pre_b1[ln];
    #pragma unroll
    for (int t = 0; t < 2; ++t) {
      long s = base + t*16 + ln;
      v2f a;
      a.x = x[s*4 + half*2 + 0];
      a.y = x[s*4 + half*2 + 1];
      v8f c;
      #pragma unroll
      for (int r = 0; r < 8; ++r) c[r] = bias;
      v8f d = wmma_f32_k4(a, b1, c);
      #pragma unroll
      for (int r = 0; r < 8; ++r) {
        float hv = d[r] > 0.f ? d[r] : 0.f;
        ldsH[wv][t][(half*8 + r)*16 + ln] = hv;   // [sampleRow][hidden]
      }
    }
  }
  __syncthreads();

  // ---------------- P2: angles = tanh(h @ pre_w2 + pre_b2), K=16 as 4 WMMA ----
  {
    float bias2 = pre_b2[ln6];
    bias2 = (ln < 6) ? bias2 : 0.f;
    // preload padded B column (16 K-values for this lane's column), unconditional loads
    float bcol[4][2];
    #pragma unroll
    for (int kk = 0; kk < 4; ++kk) {
      int k0 = kk*4 + half*2;
      float t0 = pre_w2[(k0+0)*6 + ln6];
      float t1 = pre_w2[(k0+1)*6 + ln6];
      bcol[kk][0] = (ln < 6) ? t0 : 0.f;
      bcol[kk][1] = (ln < 6) ? t1 : 0.f;
    }
    #pragma unroll
    for (int t = 0; t < 2; ++t) {
      v8f d;
      #pragma unroll
      for (int r = 0; r < 8; ++r) d[r] = bias2;
      #pragma unroll
      for (int kk = 0; kk < 4; ++kk) {
        int k0 = kk*4 + half*2;
        v2f a, b;
        a.x = ldsH[wv][t][ln*16 + k0 + 0];
        a.y = ldsH[wv][t][ln*16 + k0 + 1];
        b.x = bcol[kk][0];
        b.y = bcol[kk][1];
        d = wmma_f32_k4(a, b, d);
      }
      #pragma unroll
      for (int r = 0; r < 8; ++r) {
        if (ln < 6) ldsV[wv][t*16 + half*8 + r][ln] = tanhf(d[r]);
      }
    }
  }
  __syncthreads();

  // ---------------- P3: exact 6-qubit statevector sim, thread-per-sample -----
  {
    // Initial stage: (RY(theta)*H)|0> per qubit -> real product state.
    float u[6], v[6];
    #pragma unroll
    for (int q = 0; q < 6; ++q) {
      float th = 0.5f * ldsV[wv][lane][q];
      float cth = cosf(th), sth = sinf(th);
      const float rs2 = 0.70710678118654752440f;
      u[q] = (cth - sth) * rs2;
      v[q] = (cth + sth) * rs2;
    }
    float re[DIM], im[DIM];
    re[0] = 1.f;
    #pragma unroll
    for (int q = 0; q < 6; ++q) {
      int size = 1 << q;
      #pragma unroll
      for (int i = (1 << q) - 1; i >= 0; --i) {
        float tpr = re[i];
        re[i + size] = tpr * v[q];
        re[i]        = tpr * u[q];
      }
    }
    #pragma unroll
    for (int i = 0; i < DIM; ++i) im[i] = 0.f;

    // Fully unrolled layers: CX ladders become pure register renaming.
    #pragma unroll
    for (int layer = 0; layer < 3; ++layer) {
      #pragma unroll
      for (int cq = 0; cq < 5; ++cq) {
        #pragma unroll
        for (int j = 0; j < DIM; ++j) {
          if (((j >> cq) & 1) && !((j >> (cq+1)) & 1)) {
            int j2 = j ^ (1 << (cq+1));
            float tr = re[j]; re[j] = re[j2]; re[j2] = tr;
            float ti = im[j]; im[j] = im[j2]; im[j2] = ti;
          }
        }
      }
      #pragma unroll
      for (int q = 0; q < 6; ++q) {
        int p = layer * 18 + q * 3;
        float a2 = 0.5f * qw[p + 0];
        float b2 = 0.5f * qw[p + 1];
        float g2 = 0.5f * qw[p + 2];
        float ca = cosf(a2), sa = sinf(a2);
        float cb = cosf(b2), sb = sinf(b2);
        float cg = cosf(g2), sg = sinf(g2);
        // M = RZ * RY * RX (reference applies RX, then RY, then RZ)
        c2 RXm[2][2] = { { {ca,0.f}, {0.f,-sa} }, { {0.f,-sa}, {ca,0.f} } };
        c2 RYm[2][2] = { { {cb,0.f}, {-sb,0.f} }, { {sb,0.f}, {cb,0.f} } };
        c2 RZm[2][2] = { { {cg,-sg}, {0.f,0.f} }, { {0.f,0.f}, {cg,sg} } };
        c2 T[2][2], M[2][2];
        #pragma unroll
        for (int i = 0; i < 2; ++i)
          #pragma unroll
          for (int j = 0; j < 2; ++j)
            T[i][j] = cadd(cmul(RYm[i][0], RXm[0][j]), cmul(RYm[i][1], RXm[1][j]));
        #pragma unroll
        for (int i = 0; i < 2; ++i)
          #pragma unroll
          for (int j = 0; j < 2; ++j)
            M[i][j] = cadd(cmul(RZm[i][0], T[0][j]), cmul(RZm[i][1], T[1][j]));
        #pragma unroll
        for (int i = 0; i < DIM; ++i) {
          if (!((i >> q) & 1)) {
            int j = i | (1 << q);
            c2 s0 = { re[i], im[i] };
            c2 s1 = { re[j], im[j] };
            c2 o0 = cadd(cmul(M[0][0], s0), cmul(M[0][1], s1));
            c2 o1 = cadd(cmul(M[1][0], s0), cmul(M[1][1], s1));
            re[i] = o0.r; im[i] = o0.i;
            re[j] = o1.r; im[j] = o1.i;
          }
        }
      }
    }

    // Z expectations: e[q] = sum_i (1-2*bit_q(i)) * |amp_i|^2
    float e[6] = {0.f,0.f,0.f,0.f,0.f,0.f};
    #pragma unroll
    for (int i = 0; i < DIM; ++i) {
      float pr = re[i]*re[i] + im[i]*im[i];
      #pragma unroll
      for (int q = 0; q < 6; ++q) e[q] += ((i >> q) & 1) ? -pr : pr;
    }
    #pragma unroll
    for (int q = 0; q < 6; ++q) ldsV[wv][lane][q] = e[q];   // same slots this thread read
  }
  __syncthreads();

  // ---------------- P4: h2 = relu(exp @ post_w1 + post_b1), K=6 padded to 8 ---
  {
    float bias = post_b1[ln];
    float bcol[2][2];
    #pragma unroll
    for (int kk = 0; kk < 2; ++kk) {
      int k0 = kk*4 + half*2;
      float t0 = post_w1[imin(k0+0, 5)*16 + ln];
      float t1 = post_w1[imin(k0+1, 5)*16 + ln];
      bcol[kk][0] = (k0+0 < 6) ? t0 : 0.f;
      bcol[kk][1] = (k0+1 < 6) ? t1 : 0.f;
    }
    #pragma unroll
    for (int t = 0; t < 2; ++t) {
      v8f d;
      #pragma unroll
      for (int r = 0; r < 8; ++r) d[r] = bias;
      #pragma unroll
      for (int kk = 0; kk < 2; ++kk) {
        int k0 = kk*4 + half*2;
        v2f a, b;
        float a0 = ldsV[wv][t*16 + ln][imin(k0+0, 5)];
        float a1 = ldsV[wv][t*16 + ln][imin(k0+1, 5)];
        a.x = (k0+0 < 6) ? a0 : 0.f;
        a.y = (k0+1 < 6) ? a1 : 0.f;
        b.x = bcol[kk][0];
        b.y = bcol[kk][1];
        d = wmma_f32_k4(a, b, d);
      }
      #pragma unroll
      for (int r = 0; r < 8; ++r) {
        float hv = d[r] > 0.f ? d[r] : 0.f;
        ldsH[wv][t][(half*8 + r)*16 + ln] = hv;
      }
    }
  }
  __syncthreads();

  // ---------------- P5: out = h2 @ post_w2 + post_b2, K=16 as 4 WMMA ----------
  {
    float bias = post_b2[ln4];
    bias = (ln < 4) ? bias : 0.f;
    float bcol[4][2];
    #pragma unroll
    for (int kk = 0; kk < 4; ++kk) {
      int k0 = kk*4 + half*2;
      float t0 = post_w2[(k0+0)*4 + ln4];
      float t1 = post_w2[(k0+1)*4 + ln4];
      bcol[kk][0] = (ln < 4) ? t0 : 0.f;
      bcol[kk][1] = (ln < 4) ? t1 : 0.f;
    }
    #pragma unroll
    for (int t = 0; t < 2; ++t) {
      v8f d;
      #pragma unroll
      for (int r = 0; r < 8; ++r) d[r] = bias;
      #pragma unroll
      for (int kk = 0; kk < 4; ++kk) {
        int k0 = kk*4 + half*2;
        v2f a, b;
        a.x = ldsH[wv][t][ln*16 + k0 + 0];
        a.y = ldsH[wv][t][ln*16 + k0 + 1];
        b.x = bcol[kk][0];
        b.y = bcol[kk][1];
        d = wmma_f32_k4(a, b, d);
      }
      long sb = base + t*16;
      #pragma unroll
      for (int r = 0; r < 8; ++r) {
        if (ln < 4) out[(sb + half*8 + r)*4 + ln] = d[r];
      }
    }
  }
}

extern "C" void kernel_launch(void* const* d_in, const int* in_sizes, int n_in,
                              void* d_out, int out_size, void* d_ws, size_t ws_size,
                              hipStream_t stream) {
  (void)n_in; (void)out_size; (void)d_ws; (void)ws_size;
  const float* x       = (const float*)d_in[0];
  const float* pre_w1  = (const float*)d_in[1];
  const float* pre_b1  = (const float*)d_in[2];
  const float* pre_w2  = (const float*)d_in[3];
  const float* pre_b2  = (const float*)d_in[4];
  const float* qw      = (const float*)d_in[5];
  const float* post_w1 = (const float*)d_in[6];
  const float* post_b1 = (const float*)d_in[7];
  const float* post_w2 = (const float*)d_in[8];
  const float* post_b2 = (const float*)d_in[9];
  float* out = (float*)d_out;

  int batch  = in_sizes[0] / 4;   // FEATURE_DIM = 4
  int blocks = batch / 256;       // 256 samples per block (8 waves x 32)
  qnn_fused<<<blocks, 256, 0, stream>>>(x, pre_w1, pre_b1, pre_w2, pre_b2, qw,
                                        post_w1, post_b1, post_w2, post_b2, out);
}